// StreamingAttnLayer_27101243637999
// MI455X (gfx1250) — compile-verified
//
#include <hip/hip_runtime.h>
#include <hip/hip_bf16.h>

typedef __attribute__((ext_vector_type(2))) float v2f;
typedef __attribute__((ext_vector_type(8))) float v8f;

#define LCH 128   // chunk length
#define BB  8     // batch
#define NHH 16    // heads
#define DKK 64    // head dim
#define DM  1024  // model dim

// ---------------------------------------------------------------------------
// Generic wave-level fp32 WMMA GEMM: C[M x N] = A[M x K] @ B[K x N] + bias[N]
// grid = (N/64, M/16), block = 32 (one wave). Each wave owns a 16x64 tile,
// reusing the A fragment across four 16x16 WMMA n-subtiles.
// A-frag layout (V_WMMA_F32_16X16X4_F32): lanes 0-15 -> M rows, K=0/1;
// lanes 16-31 -> same M rows, K=2/3. B-frag mirrors with lane = N column.
// ---------------------------------------------------------------------------
__global__ __launch_bounds__(32)
void gemm_f32_wmma(const float* __restrict__ A, const float* __restrict__ Bm,
                   const float* __restrict__ bias, float* __restrict__ C,
                   int N, int K)
{
  const int lane  = threadIdx.x;
  const int col   = lane & 15;
  const int khalf = lane >> 4;          // 0: K={0,1}, 1: K={2,3}
  const int m0 = blockIdx.y << 4;
  const int n0 = blockIdx.x << 6;

  v8f acc0 = {}, acc1 = {}, acc2 = {}, acc3 = {};
  const float* arow = A + (size_t)(m0 + col) * K + 2 * khalf;
  for (int k0 = 0; k0 < K; k0 += 4) {
    v2f a = *(const v2f*)(arow + k0);                       // A[m][k], A[m][k+1]
    const float* bp = Bm + (size_t)(k0 + 2 * khalf) * N + n0 + col;
    v2f b0, b1, b2, b3;
    b0.x = bp[0];  b0.y = bp[N];
    b1.x = bp[16]; b1.y = bp[N + 16];
    b2.x = bp[32]; b2.y = bp[N + 32];
    b3.x = bp[48]; b3.y = bp[N + 48];
    acc0 = __builtin_amdgcn_wmma_f32_16x16x4_f32(false, a, false, b0, (short)0, acc0, false, false);
    acc1 = __builtin_amdgcn_wmma_f32_16x16x4_f32(false, a, false, b1, (short)0, acc1, false, false);
    acc2 = __builtin_amdgcn_wmma_f32_16x16x4_f32(false, a, false, b2, (short)0, acc2, false, false);
    acc3 = __builtin_amdgcn_wmma_f32_16x16x4_f32(false, a, false, b3, (short)0, acc3, false, false);
  }
  const int rbase = m0 + (khalf << 3);
  const float bs0 = bias[n0 + col],      bs1 = bias[n0 + 16 + col];
  const float bs2 = bias[n0 + 32 + col], bs3 = bias[n0 + 48 + col];
  #pragma unroll
  for (int r = 0; r < 8; ++r) {
    float* crow = C + (size_t)(rbase + r) * N + n0 + col;
    crow[0]  = acc0[r] + bs0;
    crow[16] = acc1[r] + bs1;
    crow[32] = acc2[r] + bs2;
    crow[48] = acc3[r] + bs3;
  }
}

// ---------------------------------------------------------------------------
// Attention: with W == L and T0 == 4096, only the new chunk's keys are in the
// window, causally. One workgroup per (b,h); 8 waves; wave w owns query rows
// [16w, 16w+16). S strip by WMMA -> LDS; per-row softmax; O = P@V by WMMA.
// ---------------------------------------------------------------------------
__global__ __launch_bounds__(256)
void attn_kernel(const float* __restrict__ qkv, float* __restrict__ obuf)
{
  __shared__ float Ss[LCH][LCH];                 // 64 KB
  const int bh = blockIdx.x;
  const int b  = bh >> 4;
  const int h  = bh & 15;
  const int tid   = threadIdx.x;
  const int wave  = tid >> 5;
  const int lane  = tid & 31;
  const int col   = lane & 15;
  const int khalf = lane >> 4;
  const int l0 = wave << 4;

  const size_t rstride = (size_t)BB * 3 * DM;    // qkv row stride for fixed b
  const float* Qb = qkv + (size_t)b * 3 * DM + h * DKK;        // q block
  const float* Kb = Qb + DM;                                   // k block
  const float* Vb = Qb + 2 * DM;                               // v block

  // ---- phase 1: S = Q @ K^T / sqrt(DK), causal mask, into LDS ----
  for (int jt = 0; jt <= wave; ++jt) {
    const int t0 = jt << 4;
    v8f acc = {};
    for (int d0 = 0; d0 < DKK; d0 += 4) {
      v2f a  = *(const v2f*)(Qb + (size_t)(l0 + col) * rstride + d0 + 2 * khalf);
      v2f bb = *(const v2f*)(Kb + (size_t)(t0 + col) * rstride + d0 + 2 * khalf);
      acc = __builtin_amdgcn_wmma_f32_16x16x4_f32(false, a, false, bb, (short)0, acc, false, false);
    }
    #pragma unroll
    for (int r = 0; r < 8; ++r) {
      const int lrow = l0 + (khalf << 3) + r;
      const int tcol = t0 + col;
      Ss[lrow][tcol] = (tcol <= lrow) ? acc[r] * 0.125f : -1e9f;
    }
  }
  __syncthreads();

  // ---- phase 2: row softmax over valid prefix [0, l] ----
  if (tid < LCH) {
    const int l = tid;
    float m = -1e30f;
    for (int t = 0; t <= l; ++t) m = fmaxf(m, Ss[l][t]);
    float sum = 0.f;
    for (int t = 0; t <= l; ++t) { float e = __expf(Ss[l][t] - m); Ss[l][t] = e; sum += e; }
    const float inv = 1.0f / sum;
    for (int t = 0; t <= l; ++t) Ss[l][t] *= inv;
    for (int t = l + 1; t < LCH; ++t) Ss[l][t] = 0.f;   // zero so P@V over full K works
  }
  __syncthreads();

  // ---- phase 3: O(16x64) = P(16x128) @ V(128x64) ----
  v8f oacc0 = {}, oacc1 = {}, oacc2 = {}, oacc3 = {};
  for (int kk = 0; kk < LCH; kk += 4) {
    v2f a = *(const v2f*)(&Ss[l0 + col][kk + 2 * khalf]);
    const float* vp0 = Vb + (size_t)(kk + 2 * khalf)     * rstride + col;
    const float* vp1 = Vb + (size_t)(kk + 2 * khalf + 1) * rstride + col;
    v2f b0, b1, b2, b3;
    b0.x = vp0[0];  b0.y = vp1[0];
    b1.x = vp0[16]; b1.y = vp1[16];
    b2.x = vp0[32]; b2.y = vp1[32];
    b3.x = vp0[48]; b3.y = vp1[48];
    oacc0 = __builtin_amdgcn_wmma_f32_16x16x4_f32(false, a, false, b0, (short)0, oacc0, false, false);
    oacc1 = __builtin_amdgcn_wmma_f32_16x16x4_f32(false, a, false, b1, (short)0, oacc1, false, false);
    oacc2 = __builtin_amdgcn_wmma_f32_16x16x4_f32(false, a, false, b2, (short)0, oacc2, false, false);
    oacc3 = __builtin_amdgcn_wmma_f32_16x16x4_f32(false, a, false, b3, (short)0, oacc3, false, false);
  }
  #pragma unroll
  for (int r = 0; r < 8; ++r) {
    const int l = l0 + (khalf << 3) + r;
    float* orow = obuf + (size_t)(l * BB + b) * DM + h * DKK + col;
    orow[0]  = oacc0[r];
    orow[16] = oacc1[r];
    orow[32] = oacc2[r];
    orow[48] = oacc3[r];
  }
}

// ---------------------------------------------------------------------------
// Bulk copy of the old KV cache into the concatenated output (float4, grid-stride)
// ---------------------------------------------------------------------------
__global__ void copy_f4(const float4* __restrict__ src, float4* __restrict__ dst, size_t n4)
{
  size_t i = (size_t)blockIdx.x * blockDim.x + threadIdx.x;
  const size_t stride = (size_t)gridDim.x * blockDim.x;
  for (; i < n4; i += stride) dst[i] = src[i];
}

// Append the new chunk's k/v (from the qkv scratch buffer) as rows [4096,4224)
__global__ __launch_bounds__(256)
void kv_tail(const float* __restrict__ qkv, float* __restrict__ outK, float* __restrict__ outV)
{
  const int idx  = blockIdx.x * blockDim.x + threadIdx.x;  // 0 .. 128*8*1024-1
  const int colv = idx & (DM - 1);
  const int lb   = idx >> 10;                              // l*B + b
  const size_t orow = (size_t)(4096 * BB + lb) * DM + colv;
  const size_t qrow = (size_t)lb * (3 * DM);
  outK[orow] = qkv[qrow + DM + colv];
  outV[orow] = qkv[qrow + 2 * DM + colv];
}

extern "C" void kernel_launch(void* const* d_in, const int* in_sizes, int n_in,
                              void* d_out, int out_size, void* d_ws, size_t ws_size,
                              hipStream_t stream)
{
  (void)in_sizes; (void)n_in; (void)out_size; (void)ws_size;
  const float* x     = (const float*)d_in[0];   // (128, 8, 1024)
  const float* Kc    = (const float*)d_in[1];   // (4096, 8, 16, 64)
  const float* Vc    = (const float*)d_in[2];
  const float* qkv_w = (const float*)d_in[3];   // (1024, 3072)
  const float* qkv_b = (const float*)d_in[4];
  const float* out_w = (const float*)d_in[5];   // (1024, 1024)
  const float* out_b = (const float*)d_in[6];

  float* y    = (float*)d_out;                         // 128*8*1024      = 1,048,576
  float* outK = y + (size_t)1048576;                   // 4224*8*16*64    = 34,603,008
  float* outV = outK + (size_t)34603008;

  // scratch: qkv (1024 x 3072) then o (1024 x 1024)  -> 16 MB of d_ws
  float* qkv  = (float*)d_ws;
  float* obuf = qkv + (size_t)1024 * 3072;

  // 1) qkv = x @ qkv_w + qkv_b   (M=1024, N=3072, K=1024)
  gemm_f32_wmma<<<dim3(3072 / 64, 1024 / 16), 32, 0, stream>>>(x, qkv_w, qkv_b, qkv, 3072, 1024);

  // 2) KV-cache concat: bulk copy old cache + append new k/v
  const size_t cache_f4 = (size_t)4096 * BB * DM / 4;  // 8,388,608 float4
  copy_f4<<<8192, 256, 0, stream>>>((const float4*)Kc, (float4*)outK, cache_f4);
  copy_f4<<<8192, 256, 0, stream>>>((const float4*)Vc, (float4*)outV, cache_f4);
  kv_tail<<<(LCH * BB * DM) / 256, 256, 0, stream>>>(qkv, outK, outV);

  // 3) windowed causal attention within the chunk
  attn_kernel<<<BB * NHH, 256, 0, stream>>>(qkv, obuf);

  // 4) y = o @ out_w + out_b   (M=1024, N=1024, K=1024)
  gemm_f32_wmma<<<dim3(1024 / 64, 1024 / 16), 32, 0, stream>>>(obuf, out_w, out_b, y, 1024, 1024);
}